// WeightedRMSELoss_25546465476984
// MI455X (gfx1250) — compile-verified
//
#include <hip/hip_runtime.h>
#include <math.h>

typedef __attribute__((ext_vector_type(2))) float v2f;
typedef __attribute__((ext_vector_type(8))) float v8f;

static constexpr int kB  = 64;
static constexpr int kH  = 512;
static constexpr int kW  = 512;
static constexpr int kHW = kH * kW;              // 262144 elements per sample
static constexpr int kBlocksPerSample = 64;      // 4096 floats (1024 float4) per block
static constexpr int kThreads = 256;             // 8 waves -> 4 float4 per thread
static constexpr int kRoiHalf = 16;

// Kernel 1: streaming weighted-SSE partial sums.
// grid = kB * kBlocksPerSample blocks; block (s*64 + chunk) covers float4 range
// [chunk*1024, (chunk+1)*1024) of sample s. Each block writes exactly one
// partial to part[blockIdx.x] (unique slot -> deterministic, no atomics, no
// need to pre-zero the poisoned workspace).
__global__ __launch_bounds__(kThreads)
void wrmse_partial_kernel(const float* __restrict__ preds,
                          const float* __restrict__ targets,
                          const float* __restrict__ rx,
                          float* __restrict__ part) {
  const int blk   = blockIdx.x;
  const int s     = blk >> 6;       // sample index
  const int chunk = blk & 63;       // chunk within sample
  const int tid   = threadIdx.x;

  // ROI box for this sample (uniform across the block; hits one cacheline).
  const int cx = (int)floorf(rx[2 * s + 0] * (float)kW);
  const int cy = (int)floorf(rx[2 * s + 1] * (float)kH);
  const int x0 = cx - kRoiHalf, x1 = cx + kRoiHalf;
  const int y0 = cy - kRoiHalf, y1 = cy + kRoiHalf;

  const float4* __restrict__ p4 =
      reinterpret_cast<const float4*>(preds + (size_t)s * kHW);
  const float4* __restrict__ t4 =
      reinterpret_cast<const float4*>(targets + (size_t)s * kHW);

  float acc = 0.0f;
#pragma unroll
  for (int k = 0; k < 4; ++k) {
    const int f4  = chunk * 1024 + k * kThreads + tid;  // float4 index in sample
    const float4 p = p4[f4];
    const float4 t = t4[f4];
    const int lin = f4 << 2;              // element index in sample
    const int row = lin >> 9;             // / 512
    const int c0  = lin & (kW - 1);       // column of .x component
    const bool iny = (row >= y0) & (row < y1);
    float d;
    d = p.x - t.x; acc += d * d * ((iny & (c0 + 0 >= x0) & (c0 + 0 < x1)) ? 10.0f : 1.0f);
    d = p.y - t.y; acc += d * d * ((iny & (c0 + 1 >= x0) & (c0 + 1 < x1)) ? 10.0f : 1.0f);
    d = p.z - t.z; acc += d * d * ((iny & (c0 + 2 >= x0) & (c0 + 2 < x1)) ? 10.0f : 1.0f);
    d = p.w - t.w; acc += d * d * ((iny & (c0 + 3 >= x0) & (c0 + 3 < x1)) ? 10.0f : 1.0f);
  }

  // Block reduction: 256 partials -> LDS -> wave 0 holds 32 partials.
  __shared__ float s_part[kThreads];
  s_part[tid] = acc;
  __syncthreads();

  if (tid < 32) {  // wave 0 only; EXEC all 1s within the wave (WMMA requirement)
    float psum = 0.0f;
#pragma unroll
    for (int i = 0; i < kThreads / 32; ++i) psum += s_part[tid + i * 32];

    // 32-lane sum via one FP32 WMMA: D(16x16) = A(16x4) * ones(4x16) + 0.
    // A layout (wave32): lane L<16 supplies A[L][0],A[L][1]; lane L>=16
    // supplies A[L-16][2],A[L-16][3]. Put this lane's partial in slot 0 and
    // zero slot 1 -> rowsum(r) = psum(lane r) + psum(lane r+16).
    v2f a; a.x = psum; a.y = 0.0f;
    v2f b; b.x = 1.0f; b.y = 1.0f;   // all-ones B regardless of its lane mapping
    v8f c = {};
    v8f dm = __builtin_amdgcn_wmma_f32_16x16x4_f32(
        /*neg_a=*/false, a, /*neg_b=*/false, b,
        /*c_mod=*/(short)0, c, /*reuse_a=*/false, /*reuse_b=*/false);

    // D layout: lanes 0-15 hold rows 0-7 in dm[0..7]; lanes 16-31 hold rows
    // 8-15. Sum own 8 rows, then fold across the half-wave boundary.
    float lanesum = dm[0] + dm[1] + dm[2] + dm[3] + dm[4] + dm[5] + dm[6] + dm[7];
    float total = lanesum + __shfl_xor(lanesum, 16, 32);
    if (tid == 0) part[blk] = total;
  }
}

// Kernel 2: finish. Thread s (0..63) sums its sample's 64 partials, takes
// sqrt(mean), then the 64 per-sample RMS values are averaged.
__global__ __launch_bounds__(64)
void wrmse_final_kernel(const float* __restrict__ part, float* __restrict__ out) {
  const int s = threadIdx.x;  // sample index 0..63 (2 waves)
  float sum = 0.0f;
#pragma unroll
  for (int b = 0; b < kBlocksPerSample; ++b) sum += part[s * kBlocksPerSample + b];
  float rms = sqrtf(sum * (1.0f / (float)kHW));
#pragma unroll
  for (int off = 16; off > 0; off >>= 1) rms += __shfl_xor(rms, off, 32);
  __shared__ float sw[2];
  if ((threadIdx.x & 31) == 0) sw[threadIdx.x >> 5] = rms;
  __syncthreads();
  if (threadIdx.x == 0) out[0] = (sw[0] + sw[1]) * (1.0f / (float)kB);
}

extern "C" void kernel_launch(void* const* d_in, const int* in_sizes, int n_in,
                              void* d_out, int out_size, void* d_ws, size_t ws_size,
                              hipStream_t stream) {
  (void)in_sizes; (void)n_in; (void)out_size; (void)ws_size;
  const float* preds   = (const float*)d_in[0];
  const float* targets = (const float*)d_in[1];
  const float* rx      = (const float*)d_in[2];
  float* out  = (float*)d_out;
  float* part = (float*)d_ws;  // kB * kBlocksPerSample = 4096 floats (16 KiB)

  wrmse_partial_kernel<<<kB * kBlocksPerSample, kThreads, 0, stream>>>(
      preds, targets, rx, part);
  wrmse_final_kernel<<<1, 64, 0, stream>>>(part, out);
}